// Model_41266045780463
// MI455X (gfx1250) — compile-verified
//
#include <hip/hip_runtime.h>
#include <hip/hip_bf16.h>

#define B_   512
#define CIN  32
#define RIN  64
#define HID  128
#define OUT_ 512
#define HH   16
#define WW   16
#define LL   (HH*WW)   // 256 sequence steps

typedef float v2f __attribute__((ext_vector_type(2)));
typedef float v8f __attribute__((ext_vector_type(8)));

// Exact-precision fp32 WMMA: D(16x16) = A(16x4) * B(4x16) + C
__device__ __forceinline__ v8f wmma_f32x4(v2f a, v2f b, v8f c) {
  return __builtin_amdgcn_wmma_f32_16x16x4_f32(false, a, false, b, (short)0, c,
                                               false, false);
}

__device__ __forceinline__ float fast_tanh(float v) {
#if __has_builtin(__builtin_amdgcn_tanhf)
  return __builtin_amdgcn_tanhf(v);   // v_tanh_f32 hardware transcendental
#else
  return tanhf(v);
#endif
}

// ---------------------------------------------------------------------------
// Kernel 1: fuse W_c = W_ih @ W_in (128x32), b_c = W_ih@b_in + b_ih + b_hh
// ---------------------------------------------------------------------------
__global__ void fuse_weights_kernel(const float* __restrict__ W_in,
                                    const float* __restrict__ b_in,
                                    const float* __restrict__ W_ih,
                                    const float* __restrict__ b_ih,
                                    const float* __restrict__ b_hh,
                                    float* __restrict__ W_c,
                                    float* __restrict__ b_c) {
  const int i = threadIdx.x;  // 0..127
  float wc[CIN];
#pragma unroll
  for (int k = 0; k < CIN; ++k) wc[k] = 0.f;
  float bc = b_ih[i] + b_hh[i];
  for (int j = 0; j < RIN; ++j) {
    const float wij = W_ih[i * RIN + j];
    bc += wij * b_in[j];
#pragma unroll
    for (int k = 0; k < CIN; ++k) wc[k] += wij * W_in[j * CIN + k];
  }
#pragma unroll
  for (int k = 0; k < CIN; ++k) W_c[i * CIN + k] = wc[k];
  b_c[i] = bc;
}

// ---------------------------------------------------------------------------
// Kernel 2: fused input-projection + RNN scan.
// 32 blocks (16 batch rows each) x 8 waves (one 16-col tile each).
// Weight B-fragments are register-resident. Per step:
//   (a) issue all 32 h A-fragment ds_load_b64 (independent -> clause),
//   (b) 8 x-part WMMAs from registers while those loads fly,
//   (c) 32 h-part WMMAs draining the loads with partial dscnt waits,
//   (d) v_tanh -> LDS ping-pong + global history, one barrier.
// ---------------------------------------------------------------------------
__global__ void __launch_bounds__(256)
rnn_scan_kernel(const float* __restrict__ x,     // (B, CIN, 256)
                const float* __restrict__ h0,    // (B, HID)
                const float* __restrict__ W_hh,  // (HID, HID)
                const float* __restrict__ W_c,   // (HID, CIN)
                const float* __restrict__ b_c,   // (HID)
                float* __restrict__ hs) {        // (L, B, HID)
  extern __shared__ float smem[];
  float* s_h = smem;  // 2 * (16*HID) ping-pong

  const int tid  = threadIdx.x;
  const int lane = tid & 31;
  const int wid  = tid >> 5;  // 0..7, owns col tile `wid`
  const int b0   = blockIdx.x * 16;

  // ISA fragment lane decomposition (wave32):
  const int mlo   = lane & 15;         // A row / B col / C col within tile
  const int khalf = (lane >> 4) << 1;  // K-pair select: 0 or 2
  const int mhi   = (lane >> 4) << 3;  // C/D row base: 0 or 8
  const int n     = wid * 16 + mlo;    // global hidden column for this lane

  // Register-resident loop-invariant B-fragments (B[k,n] = W[n,k]):
  v2f bhh[HID / 4];  // 32 x v2f = 64 VGPRs
  {
    const v2f* wrow = (const v2f*)(W_hh + (size_t)n * HID);
#pragma unroll
    for (int k2 = 0; k2 < HID / 4; ++k2) bhh[k2] = wrow[k2 * 2 + (khalf >> 1)];
  }
  v2f bwc[CIN / 4];  // 8 x v2f = 16 VGPRs
  {
    const v2f* wrow = (const v2f*)(W_c + (size_t)n * CIN);
#pragma unroll
    for (int k2 = 0; k2 < CIN / 4; ++k2) bwc[k2] = wrow[k2 * 2 + (khalf >> 1)];
  }
  const float bias = b_c[n];

  // h0 -> ping-pong buffer 0
  for (int i = tid; i < 16 * HID; i += blockDim.x)
    s_h[i] = h0[(size_t)(b0 + (i >> 7)) * HID + (i & (HID - 1))];
  __syncthreads();

  // x A-operand prefetch (double-buffered): x_seq[b, l, kk] = x[b, kk, l]
  const float* xbase = x + (size_t)(b0 + mlo) * (CIN * LL);
  v2f xa[CIN / 4];
#pragma unroll
  for (int k2 = 0; k2 < CIN / 4; ++k2) {
    const int kk = k2 * 4 + khalf;
    xa[k2].x = xbase[(size_t)kk * LL];
    xa[k2].y = xbase[(size_t)(kk + 1) * LL];
  }

  for (int l = 0; l < LL; ++l) {
    const int rb = l & 1;
    const float* hprev = s_h + rb * (16 * HID);
    float* hnext = s_h + (rb ^ 1) * (16 * HID);

    // (a) issue ALL h A-fragment loads up front (independent ds_load_b64s)
    const float* hrow = hprev + mlo * HID;
    v2f ha[HID / 4];
#pragma unroll
    for (int k2 = 0; k2 < HID / 4; ++k2)
      ha[k2] = *(const v2f*)(hrow + k2 * 4 + khalf);

    // prefetch next step's x operands (global loads overlap the WMMA chain)
    v2f xn[CIN / 4];
    if (l + 1 < LL) {
      const float* xc = xbase + (l + 1);
#pragma unroll
      for (int k2 = 0; k2 < CIN / 4; ++k2) {
        const int kk = k2 * 4 + khalf;
        xn[k2].x = xc[(size_t)kk * LL];
        xn[k2].y = xc[(size_t)(kk + 1) * LL];
      }
    }

    v8f acc;
#pragma unroll
    for (int r = 0; r < 8; ++r) acc[r] = bias;

    // (b) input projection first: operands already in registers
#pragma unroll
    for (int k2 = 0; k2 < CIN / 4; ++k2) acc = wmma_f32x4(xa[k2], bwc[k2], acc);

    // (c) recurrent part: drain the preloaded A-fragments
#pragma unroll
    for (int k2 = 0; k2 < HID / 4; ++k2) acc = wmma_f32x4(ha[k2], bhh[k2], acc);

    // (d) tanh -> ping-pong LDS (next step's A operand) + global hs history
    float* outl = hs + (size_t)l * (B_ * HID);
#pragma unroll
    for (int r = 0; r < 8; ++r) {
      const int m = mhi + r;
      const float hv = fast_tanh(acc[r]);
      hnext[m * HID + n] = hv;
      outl[(size_t)(b0 + m) * HID + n] = hv;
    }

#pragma unroll
    for (int k2 = 0; k2 < CIN / 4; ++k2) xa[k2] = xn[k2];
    __syncthreads();
  }
}

// ---------------------------------------------------------------------------
// Kernel 3: bilinear upsample (align_corners) 16x16 -> 32x32 + hardswish +
// spatial mean. One thread per (b, c); hs (64 MB) is L2-resident (192 MB L2)
// so row re-reads are cheap. ox loop fully unrolled -> taps are compile-time.
// ---------------------------------------------------------------------------
__global__ void upsample_pool_kernel(const float* __restrict__ hs,
                                     float* __restrict__ pooled) {
  const int tid = blockIdx.x * blockDim.x + threadIdx.x;  // 0..B*HID-1
  const int b = tid >> 7;
  const int c = tid & (HID - 1);
  const float* base = hs + (size_t)b * HID + c;  // element l at base[l*B*HID]
  float acc = 0.f;
  for (int oy = 0; oy < 2 * HH; ++oy) {
    const float ysf = (oy * 15.0f) / 31.0f;
    const int y0 = (int)ysf;
    const int y1 = (y0 + 1 < HH) ? y0 + 1 : HH - 1;
    const float wy = ysf - (float)y0;
    float r0[WW], r1[WW];
#pragma unroll
    for (int xx = 0; xx < WW; ++xx) {
      r0[xx] = base[(size_t)(y0 * WW + xx) * (B_ * HID)];
      r1[xx] = base[(size_t)(y1 * WW + xx) * (B_ * HID)];
    }
#pragma unroll
    for (int ox = 0; ox < 2 * WW; ++ox) {
      const float xsf = (ox * 15.0f) / 31.0f;  // folded at compile time
      const int x0 = (int)xsf;
      const int x1 = (x0 + 1 < WW) ? x0 + 1 : WW - 1;
      const float wx = xsf - (float)x0;
      const float v0 = r0[x0] + wx * (r0[x1] - r0[x0]);
      const float v1 = r1[x0] + wx * (r1[x1] - r1[x0]);
      const float v = v0 + wy * (v1 - v0);
      acc += v * fminf(fmaxf(v + 3.0f, 0.0f), 6.0f) * (1.0f / 6.0f);
    }
  }
  pooled[tid] = acc * (1.0f / (4.0f * HH * WW));
}

// ---------------------------------------------------------------------------
// Kernel 4: out = pooled(512x128) @ W_out^T(128x512) + b_out.
// 1024 16x16 tiles, one wave each, K=128 -> 32 fp32 WMMA per wave.
// ---------------------------------------------------------------------------
__global__ void __launch_bounds__(128)
out_gemm_kernel(const float* __restrict__ pooled,
                const float* __restrict__ W_out,
                const float* __restrict__ b_out,
                float* __restrict__ out) {
  const int lane = threadIdx.x & 31;
  const int gw = (blockIdx.x * blockDim.x + threadIdx.x) >> 5;  // 0..1023
  const int b0 = (gw >> 5) * 16;
  const int o0 = (gw & 31) * 16;
  const int mlo = lane & 15;
  const int khalf = (lane >> 4) << 1;
  const int mhi = (lane >> 4) << 3;

  const int n = o0 + mlo;
  v8f acc;
  const float bias = b_out[n];
#pragma unroll
  for (int r = 0; r < 8; ++r) acc[r] = bias;

  const v2f* arow = (const v2f*)(pooled + (size_t)(b0 + mlo) * HID);
  const v2f* brow = (const v2f*)(W_out + (size_t)n * HID);  // B[k,n]=W_out[n,k]
#pragma unroll
  for (int k2 = 0; k2 < HID / 4; ++k2) {
    const int idx = k2 * 2 + (khalf >> 1);
    acc = wmma_f32x4(arow[idx], brow[idx], acc);
  }
#pragma unroll
  for (int r = 0; r < 8; ++r)
    out[(size_t)(b0 + mhi + r) * OUT_ + n] = acc[r];
}

// ---------------------------------------------------------------------------
extern "C" void kernel_launch(void* const* d_in, const int* in_sizes, int n_in,
                              void* d_out, int out_size, void* d_ws,
                              size_t ws_size, hipStream_t stream) {
  const float* x     = (const float*)d_in[0];
  const float* h0    = (const float*)d_in[1];
  const float* W_in  = (const float*)d_in[2];
  const float* b_in  = (const float*)d_in[3];
  const float* W_ih  = (const float*)d_in[4];
  const float* b_ih  = (const float*)d_in[5];
  const float* W_hh  = (const float*)d_in[6];
  const float* b_hh  = (const float*)d_in[7];
  const float* W_out = (const float*)d_in[8];
  const float* b_out = (const float*)d_in[9];
  float* out = (float*)d_out;

  // Workspace (floats): W_c[4096] b_c[128] pad | hs[L*B*HID] | pooled[B*HID]
  // Total ~64.3 MB.
  float* ws = (float*)d_ws;
  float* W_c = ws;
  float* b_c = ws + HID * CIN;
  float* hs = ws + 8192;
  float* pooled = hs + (size_t)LL * B_ * HID;

  fuse_weights_kernel<<<1, HID, 0, stream>>>(W_in, b_in, W_ih, b_ih, b_hh, W_c,
                                             b_c);

  const size_t smem_bytes = (size_t)(2 * 16 * HID) * sizeof(float);
  rnn_scan_kernel<<<B_ / 16, 256, smem_bytes, stream>>>(x, h0, W_hh, W_c, b_c,
                                                        hs);

  upsample_pool_kernel<<<(B_ * HID) / 256, 256, 0, stream>>>(hs, pooled);

  out_gemm_kernel<<<256, 128, 0, stream>>>(pooled, W_out, b_out, out);
}